// EventGenerator_89498528514525
// MI455X (gfx1250) — compile-verified
//
#include <hip/hip_runtime.h>

// ---------------------------------------------------------------------------
// EventGenerator for MI455X (gfx1250). fp32 end-to-end.
//  * GEMMs via V_WMMA_F32_16X16X4_F32 (workload is memory-bound -> keep f32;
//    res_bank B-matrix (16MB) lives in the 192MB L2 across M-tile passes).
//  * fft_convolve via 32768-pt complex FFT (packed real trick) entirely in
//    LDS (256KB of the 320KB WGP LDS), one signal per workgroup. Global
//    reads are contiguous; the bit-reversal permutation is an LDS scatter.
// Workspace requirement: ~152 MB (see WS_* offsets below).
// ---------------------------------------------------------------------------

#define NEVENTS   64
#define LATENT    256
#define NRES      128
#define EXPRN     4
#define NSAMP     32768
#define ENVSZ     4096
#define NOISELEN  8192
#define NFRAMES   128
#define NGAUSS    8
#define NROOMS    8
#define FFTN      32768       // complex points (covers 65536-pt real FFT)
#define LOG2FFT   15

typedef float v2f __attribute__((ext_vector_type(2)));
typedef float v8f __attribute__((ext_vector_type(8)));

__device__ __forceinline__ int   imin(int a, int b)   { return a < b ? a : b; }
__device__ __forceinline__ int   imax(int a, int b)   { return a > b ? a : b; }
__device__ __forceinline__ float2 cmul(float2 a, float2 b) {
  return make_float2(a.x*b.x - a.y*b.y, a.x*b.y + a.y*b.x);
}
__device__ __forceinline__ unsigned br15(unsigned x) { return __brev(x) >> (32 - LOG2FFT); }

// ---------------------------------------------------------------------------
// Small linear: one block (256 thr) per (event, out) pair; K == 256 == LATENT
// ---------------------------------------------------------------------------
__global__ __launch_bounds__(256) void lin_small_kernel(
    const float* __restrict__ x, const float* __restrict__ W,
    const float* __restrict__ b, float* __restrict__ out, int nout)
{
  __shared__ float red[256];
  int bid = blockIdx.x;
  int e = bid / nout, n = bid % nout;
  int t = threadIdx.x;
  red[t] = x[e * LATENT + t] * W[n * LATENT + t];
  __syncthreads();
  for (int s = 128; s > 0; s >>= 1) {
    if (t < s) red[t] += red[t + s];
    __syncthreads();
  }
  if (t == 0) out[e * nout + n] = red[0] + b[n];
}

// ---------------------------------------------------------------------------
// WMMA f32 GEMM: C[M,N] = A[M,K] * B (+bias, optional relu).
// Compile-time flags keep the K-loop branch-free; K unrolled x4 -> 4 WMMAs
// per iteration. One 16x16 output tile per wave.
// BTRANS=1 -> B stored as [N,K] (torch weight); else row-major [K,N].
// ---------------------------------------------------------------------------
template <int BTRANS, int HASBIAS, int RELU>
__global__ __launch_bounds__(256) void gemm_wmma_kernel(
    const float* __restrict__ A, const float* __restrict__ B,
    float* __restrict__ C, const float* __restrict__ bias,
    int M, int N, int K)
{
  int lane = threadIdx.x & 31;
  int wave = (int)((blockIdx.x * (long)blockDim.x + threadIdx.x) >> 5);
  int tilesN = N >> 4;
  int tilesM = M >> 4;
  if (wave >= tilesM * tilesN) return;
  int tn = wave % tilesN, tm = wave / tilesN;
  int half = lane >> 4;          // 0: K pair {0,1}, 1: K pair {2,3}
  int l16  = lane & 15;
  int row  = tm * 16 + l16;      // A row (M)
  int col  = tn * 16 + l16;      // B/C col (N)

  const float* arow = A + (long)row * K;           // 8B-aligned at even k
  const float* bcol = BTRANS ? (B + (long)col * K) : (B + col);

  v8f acc = {0.f, 0.f, 0.f, 0.f, 0.f, 0.f, 0.f, 0.f};
#pragma unroll 4
  for (int k0 = 0; k0 < K; k0 += 4) {
    int ka = k0 + (half << 1);
    v2f a = *(const v2f*)(arow + ka);
    v2f b;
    if (BTRANS) {
      b = *(const v2f*)(bcol + ka);
    } else {
      b.x = bcol[(long)ka * N];
      b.y = bcol[(long)(ka + 1) * N];
      // stream hint for the (L2-resident) B matrix a few rows ahead
      __builtin_prefetch(bcol + (long)(ka + 8) * N, 0, 1);
    }
    acc = __builtin_amdgcn_wmma_f32_16x16x4_f32(
        false, a, false, b, (short)0, acc, false, false);
  }
  float bb = HASBIAS ? bias[col] : 0.f;
#pragma unroll
  for (int v = 0; v < 8; ++v) {
    int r = tm * 16 + v + (half ? 8 : 0);
    float val = acc[v] + bb;
    if (RELU) val = fmaxf(val, 0.f);
    C[(long)r * N + col] = val;
  }
}

// ---------------------------------------------------------------------------
// Damped harmonic oscillator bank: res_bank[128][32768]
// o1 -> o2 -> o3 chain fused per sample (tension modulation is sample-local).
// ---------------------------------------------------------------------------
__device__ __forceinline__ float dho_eval(float mraw, float draw, float ten,
                                          float x0, float amp, float t)
{
  float ms   = 2.0f  / (1.0f + __expf(-mraw));
  float dp   = 30.0f / (1.0f + __expf(-draw));
  float w0sq = __expf(ten * 2.302585092994046f) / ms;   // 10^ten / mass
  float g    = dp / (2.0f * ms);
  float wd   = sqrtf(fmaxf(w0sq - g * g, 1e-8f));
  float v    = __expf(-g * t) * (x0 * cosf(wd * t) + (g * x0) / wd * sinf(wd * t));
  return tanhf(v * amp);
}

__global__ __launch_bounds__(256) void dho_kernel(
    const float* __restrict__ m1, const float* __restrict__ d1,
    const float* __restrict__ t1, const float* __restrict__ x01,
    const float* __restrict__ a1,
    const float* __restrict__ m2, const float* __restrict__ d2,
    const float* __restrict__ t2, const float* __restrict__ x02,
    const float* __restrict__ a2,
    const float* __restrict__ infl, const float* __restrict__ infl2,
    const float* __restrict__ mix, float* __restrict__ bank)
{
  long gid = blockIdx.x * (long)blockDim.x + threadIdx.x;
  if (gid >= (long)NRES * NSAMP) return;
  int s = (int)(gid & (NSAMP - 1));
  int r = (int)(gid >> 15);
  float t = 10.0f * (float)s / (float)(NSAMP - 1);

  float w0 = mix[0], w1 = mix[1], w2 = mix[2];
  float mw = fmaxf(w0, fmaxf(w1, w2));
  float e0 = __expf(w0 - mw), e1 = __expf(w1 - mw), e2 = __expf(w2 - mw);
  float inv = 1.0f / (e0 + e1 + e2);

  int iA = r, iB = NRES + r;   // osc 0 / osc 1 parameter indices

  float o1 = dho_eval(m1[iA], d1[iA], t1[iA], x01[iA], a1[iA], t)
           + dho_eval(m1[iB], d1[iB], t1[iB], x01[iB], a1[iB], t);
  float o2 = dho_eval(m2[iA], d2[iA], t2[iA] + o1 * infl[iA],  x02[iA], a2[iA], t)
           + dho_eval(m2[iB], d2[iB], t2[iB] + o1 * infl[iB],  x02[iB], a2[iB], t);
  float o3 = dho_eval(m2[iA], d2[iA], t2[iA] + o2 * infl2[iA], x02[iA], a2[iA], t)
           + dho_eval(m2[iB], d2[iB], t2[iB] + o2 * infl2[iB], x02[iB], a2[iB], t);

  bank[gid] = (e0 * o1 + e1 * o2 + e2 * o3) * inv;
}

// ---------------------------------------------------------------------------
// Deform path: ddef(64x128) -> (64,4,32)@items(32,128) -> cumsum -> softmax(EXPR)
// ---------------------------------------------------------------------------
__global__ __launch_bounds__(128) void deform_post_kernel(
    const float* __restrict__ ddef, const float* __restrict__ items,
    float* __restrict__ dsm)
{
  __shared__ float v[EXPRN][NFRAMES];
  int e = blockIdx.x, f = threadIdx.x;
#pragma unroll
  for (int x = 0; x < EXPRN; ++x) {
    float s = 0.f;
    for (int j = 0; j < 32; ++j)
      s += ddef[e * 128 + x * 32 + j] * items[j * NFRAMES + f];
    v[x][f] = s;
  }
  __syncthreads();
  if (f < EXPRN) {                       // sequential cumsum over frames
    float c = 0.f;
    for (int i = 0; i < NFRAMES; ++i) { c += v[f][i]; v[f][i] = c; }
  }
  __syncthreads();
  float mx = fmaxf(fmaxf(v[0][f], v[1][f]), fmaxf(v[2][f], v[3][f]));
  float ex[EXPRN]; float sum = 0.f;
#pragma unroll
  for (int x = 0; x < EXPRN; ++x) { ex[x] = __expf(v[x][f] - mx); sum += ex[x]; }
  float inv = 1.0f / sum;
#pragma unroll
  for (int x = 0; x < EXPRN; ++x)
    dsm[e * (EXPRN * NFRAMES) + x * NFRAMES + f] = ex[x] * inv;
}

// ---------------------------------------------------------------------------
// Attack envelopes: gamma pdf shifted by hierarchical-dirac index (delta ->
// pure shift, no FFT), summed over gaussians, unit-norm, interp 4096->8192,
// times noise, zero-padded to 32768.  One block per event.
// ---------------------------------------------------------------------------
__global__ __launch_bounds__(256) void envelope_kernel(
    const float* __restrict__ envm, const float* __restrict__ envs,
    const float* __restrict__ enva, const float* __restrict__ pos,
    const float* __restrict__ oamp, const float* __restrict__ noise,
    float* __restrict__ ntime)
{
  __shared__ float env[ENVSZ];
  __shared__ float red[256];
  __shared__ float gk[NGAUSS], gb[NGAUSS], gc[NGAUSS], ga[NGAUSS];
  __shared__ int   gix[NGAUSS];
  __shared__ float scl;
  int e = blockIdx.x, t = threadIdx.x;

  if (t < NGAUSS) {
    float m  = fabsf(envm[e * NGAUSS + t]) + 1e-8f;
    float sd = fabsf(envs[e * NGAUSS + t]) + 1e-8f;
    float k  = (m / sd) * (m / sd);
    float th = sd * sd / m;
    gk[t] = k - 1.0f;
    gb[t] = 1.0f / th;
    gc[t] = -lgammaf(k) - k * __logf(th);
    ga[t] = fabsf(enva[e * NGAUSS + t]);
    int idx = 0;
    for (int l = 0; l < 12; ++l) {
      float p0 = pos[e * 192 + t * 24 + l * 2 + 0];
      float p1 = pos[e * 192 + t * 24 + l * 2 + 1];
      idx = (idx << 1) | (p1 > p0 ? 1 : 0);
    }
    gix[t] = idx;
  }
  __syncthreads();

  const float epsv = 1e-8f;
  const float dt   = (1.0f - epsv) / (float)(ENVSZ - 1);
  for (int i = t; i < ENVSZ; i += 256) {
    float sum = 0.f;
#pragma unroll
    for (int g = 0; g < NGAUSS; ++g) {
      int j = i - gix[g];
      if (j >= 0) {
        float tv = epsv + j * dt;
        sum += ga[g] * __expf(gk[g] * __logf(tv) - tv * gb[g] + gc[g]);
      }
    }
    env[i] = sum;
  }
  __syncthreads();

  float ps = 0.f;
  for (int i = t; i < ENVSZ; i += 256) ps += env[i] * env[i];
  red[t] = ps; __syncthreads();
  for (int s = 128; s > 0; s >>= 1) {
    if (t < s) red[t] += red[t + s];
    __syncthreads();
  }
  if (t == 0) scl = oamp[e] / (sqrtf(red[0]) + 1e-8f);
  __syncthreads();

  for (int j = t; j < NSAMP; j += 256) {
    float val = 0.f;
    if (j < NOISELEN) {                       // 4096 -> 8192 half-pixel lerp
      float sp = 0.5f * j - 0.25f;
      float ff = floorf(sp);
      float w  = sp - ff;
      int f0 = imax(0, imin(ENVSZ - 1, (int)ff));
      int f1 = imax(0, imin(ENVSZ - 1, (int)ff + 1));
      float ev = env[f0] * (1.f - w) + env[f1] * w;
      val = ev * scl * noise[e * NOISELEN + j];
    }
    ntime[e * NSAMP + j] = val;
  }
}

// ---------------------------------------------------------------------------
// In-place radix-2 DIT complex FFT over LDS (bit-reversed input order).
// ---------------------------------------------------------------------------
__device__ void fft_inplace(float2* Z, int tid, int nth)
{
  for (int len = 2; len <= FFTN; len <<= 1) {
    int half = len >> 1;
    float anginc = -6.28318530717958647692f / (float)len;
    for (int j = tid; j < (FFTN >> 1); j += nth) {
      int off = j & (half - 1);
      int i0  = ((j & ~(half - 1)) << 1) | off;
      int i1  = i0 + half;
      float sn, cs;
      __sincosf(anginc * (float)off, &sn, &cs);
      float2 u = Z[i0], v = Z[i1];
      float2 vw = make_float2(v.x * cs - v.y * sn, v.x * sn + v.y * cs);
      Z[i0] = make_float2(u.x + vw.x, u.y + vw.y);
      Z[i1] = make_float2(u.x - vw.x, u.y - vw.y);
    }
    __syncthreads();
  }
}

// Forward packed real FFT of one signal per block: real[2N] (zero-padded past
// in_len) -> Z[k] = FFT_N(x[2n] + i x[2n+1]) written to global.
// Global reads are contiguous (b64); the bit-reversal is an LDS scatter.
__global__ __launch_bounds__(1024) void fft_fwd_kernel(
    const float* __restrict__ in, long in_stride, int in_len,
    float2* __restrict__ spec, long spec_stride)
{
  extern __shared__ float2 Z[];
  const float* src = in + (long)blockIdx.x * in_stride;
  for (int n = threadIdx.x; n < FFTN; n += blockDim.x) {
    int i0 = n << 1, i1 = i0 + 1;
    float re = (i0 < in_len) ? src[i0] : 0.f;
    float im = (i1 < in_len) ? src[i1] : 0.f;
    Z[br15((unsigned)n)] = make_float2(re, im);
  }
  __syncthreads();
  fft_inplace(Z, threadIdx.x, blockDim.x);
  float2* dst = spec + (long)blockIdx.x * spec_stride;
  for (int t = threadIdx.x; t < FFTN; t += blockDim.x) dst[t] = Z[t];
}

// Unpack both packed spectra -> true 65536-pt spectra, multiply, repack,
// inverse FFT (conj/FFT/conj), emit first 32768 real samples.
__global__ __launch_bounds__(1024) void fft_mulinv_kernel(
    const float2* __restrict__ specA, long strideA,
    const float2* __restrict__ specB, long strideB, int bdiv,
    float* __restrict__ out, long out_stride)
{
  extern __shared__ float2 Z[];
  int p = blockIdx.x;
  const float2* A = specA + (long)p * strideA;
  const float2* B = specB + (long)(p / bdiv) * strideB;

  for (int k = threadIdx.x; k < FFTN; k += blockDim.x) {
    int km = (FFTN - k) & (FFTN - 1);
    float2 za = A[k], zam = A[km], zb = B[k], zbm = B[km];
    // E = (Z + conj(Zm))/2 ;  O = -i (Z - conj(Zm))/2
    float2 Ea = make_float2(0.5f * (za.x + zam.x), 0.5f * (za.y - zam.y));
    float2 Da = make_float2(0.5f * (za.x - zam.x), 0.5f * (za.y + zam.y));
    float2 Oa = make_float2(Da.y, -Da.x);
    float2 Eb = make_float2(0.5f * (zb.x + zbm.x), 0.5f * (zb.y - zbm.y));
    float2 Db = make_float2(0.5f * (zb.x - zbm.x), 0.5f * (zb.y + zbm.y));
    float2 Ob = make_float2(Db.y, -Db.x);
    float sn, cs;                                  // W = e^{-i pi k / N}
    __sincosf(-3.14159265358979323846f * (float)k / (float)FFTN, &sn, &cs);
    float2 W  = make_float2(cs, sn);
    float2 WOa = cmul(W, Oa), WOb = cmul(W, Ob);
    float2 XaP = make_float2(Ea.x + WOa.x, Ea.y + WOa.y);
    float2 XaM = make_float2(Ea.x - WOa.x, Ea.y - WOa.y);
    float2 XbP = make_float2(Eb.x + WOb.x, Eb.y + WOb.y);
    float2 XbM = make_float2(Eb.x - WOb.x, Eb.y - WOb.y);
    float2 P  = cmul(XaP, XbP);    // Y[k]
    float2 Mm = cmul(XaM, XbM);    // conj(Y[N-k])
    float2 Ep = make_float2(0.5f * (P.x + Mm.x), 0.5f * (P.y + Mm.y));
    float2 Dp = make_float2(0.5f * (P.x - Mm.x), 0.5f * (P.y - Mm.y));
    // O' = conj(W) * Dp
    float2 Op = make_float2(Dp.x * cs + Dp.y * sn, Dp.y * cs - Dp.x * sn);
    float2 Zp = make_float2(Ep.x - Op.y, Ep.y + Op.x);
    Z[br15((unsigned)k)] = make_float2(Zp.x, -Zp.y);   // conj for inverse
  }
  __syncthreads();
  fft_inplace(Z, threadIdx.x, blockDim.x);
  const float invN = 1.0f / (float)FFTN;
  float* dst = out + (long)p * out_stride;
  for (int t = threadIdx.x; t < (FFTN >> 1); t += blockDim.x) {
    dst[2 * t]     =  Z[t].x * invN;   // y = conj(FFT(conj(Z')))/N
    dst[2 * t + 1] = -Z[t].y * invN;
  }
}

// ---------------------------------------------------------------------------
// Mix deform-weighted resonances with envelope noise, tanh loudness.
// ---------------------------------------------------------------------------
__global__ __launch_bounds__(256) void combine_kernel(
    const float* __restrict__ resconv, const float* __restrict__ dsm,
    const float* __restrict__ ntime, const float* __restrict__ mxlog,
    const float* __restrict__ loud, float* __restrict__ xtime)
{
  int gid = blockIdx.x * blockDim.x + threadIdx.x;
  int e = gid >> 15, s = gid & (NSAMP - 1);
  float m0 = mxlog[e * 2], m1 = mxlog[e * 2 + 1];
  float mm = fmaxf(m0, m1);
  float e0 = __expf(m0 - mm), e1 = __expf(m1 - mm);
  float inv = 1.0f / (e0 + e1);
  // interp softmaxed deform weights 128 -> 32768 (half-pixel)
  float sp = (s + 0.5f) * (1.0f / 256.0f) - 0.5f;
  float ff = floorf(sp);
  float w  = sp - ff;
  int f0 = imax(0, imin(NFRAMES - 1, (int)ff));
  int f1 = imax(0, imin(NFRAMES - 1, (int)ff + 1));
  float xs = 0.f;
#pragma unroll
  for (int x = 0; x < EXPRN; ++x) {
    const float* dr = dsm + e * (EXPRN * NFRAMES) + x * NFRAMES;
    float dv = dr[f0] * (1.f - w) + dr[f1] * w;
    xs += resconv[(long)((e << 2) + x) * NSAMP + s] * dv;
  }
  float v = ntime[gid] * (e0 * inv) + xs * (e1 * inv);
  xtime[gid] = tanhf(v * loud[e]);
}

// Per-event room impulse: softmax(room logits) @ rooms
__global__ __launch_bounds__(256) void impulse_kernel(
    const float* __restrict__ rmlog, const float* __restrict__ rooms,
    float* __restrict__ imp)
{
  int gid = blockIdx.x * blockDim.x + threadIdx.x;
  int e = gid >> 15, s = gid & (NSAMP - 1);
  float lg[NROOMS], mx = -1e30f;
#pragma unroll
  for (int r = 0; r < NROOMS; ++r) { lg[r] = rmlog[e * NROOMS + r]; mx = fmaxf(mx, lg[r]); }
  float sum = 0.f;
#pragma unroll
  for (int r = 0; r < NROOMS; ++r) { lg[r] = __expf(lg[r] - mx); sum += lg[r]; }
  float inv = 1.0f / sum, acc = 0.f;
#pragma unroll
  for (int r = 0; r < NROOMS; ++r) acc += lg[r] * rooms[r * NSAMP + s];
  imp[gid] = acc * inv;
}

// Dry/wet blend -> final output
__global__ __launch_bounds__(256) void final_kernel(
    const float* __restrict__ xtime, const float* __restrict__ wet,
    const float* __restrict__ dwlog, float* __restrict__ out)
{
  int gid = blockIdx.x * blockDim.x + threadIdx.x;
  int e = gid >> 15;
  float m0 = dwlog[e * 2], m1 = dwlog[e * 2 + 1];
  float mm = fmaxf(m0, m1);
  float e0 = __expf(m0 - mm), e1 = __expf(m1 - mm);
  float inv = 1.0f / (e0 + e1);
  out[gid] = xtime[gid] * (e0 * inv) + wet[gid] * (e1 * inv);
}

// ---------------------------------------------------------------------------
// Workspace layout (float offsets). Regions reused across phases.
// ---------------------------------------------------------------------------
static const size_t WS_BANK  = 0;                              // 128*32768   (later: WET)
static const size_t WS_RESON = WS_BANK  + (size_t)NRES * NSAMP;          // 256*32768 (later: IMP in first 64*32768)
static const size_t WS_NTIME = WS_RESON + (size_t)NEVENTS * EXPRN * NSAMP; // 64*32768
static const size_t WS_SPECN = WS_NTIME + (size_t)NEVENTS * NSAMP;         // 64*65536
static const size_t WS_SPECR = WS_SPECN + (size_t)NEVENTS * FFTN * 2;      // 256*65536 (later: SPECX/SPECI)
static const size_t WS_XTIME = WS_SPECR + (size_t)NEVENTS * EXPRN * FFTN * 2; // 64*32768
static const size_t WS_XRES  = WS_XTIME + (size_t)NEVENTS * NSAMP;         // 64*512
static const size_t WS_POS   = WS_XRES  + (size_t)NEVENTS * 512;           // 64*192
static const size_t WS_DDEF  = WS_POS   + (size_t)NEVENTS * 192;           // 64*128
static const size_t WS_DSM   = WS_DDEF  + (size_t)NEVENTS * 128;           // 64*4*128
static const size_t WS_ENVM  = WS_DSM   + (size_t)NEVENTS * EXPRN * NFRAMES;
static const size_t WS_ENVS  = WS_ENVM  + (size_t)NEVENTS * NGAUSS;
static const size_t WS_ENVA  = WS_ENVS  + (size_t)NEVENTS * NGAUSS;
static const size_t WS_OA    = WS_ENVA  + (size_t)NEVENTS * NGAUSS;        // 64
static const size_t WS_RM    = WS_OA    + (size_t)NEVENTS;                 // 64*8
static const size_t WS_DW    = WS_RM    + (size_t)NEVENTS * NROOMS;        // 64*2
static const size_t WS_MX    = WS_DW    + (size_t)NEVENTS * 2;             // 64*2
static const size_t WS_LOUD  = WS_MX    + (size_t)NEVENTS * 2;             // 64

extern "C" void kernel_launch(void* const* d_in, const int* in_sizes, int n_in,
                              void* d_out, int out_size, void* d_ws, size_t ws_size,
                              hipStream_t stream)
{
  (void)in_sizes; (void)n_in; (void)out_size; (void)ws_size;
  const float* events  = (const float*)d_in[0];
  const float* noise   = (const float*)d_in[1];
  const float* w_tores = (const float*)d_in[2];
  const float* b_tores = (const float*)d_in[3];
  const float* w_envm  = (const float*)d_in[4];
  const float* b_envm  = (const float*)d_in[5];
  const float* w_envs  = (const float*)d_in[6];
  const float* b_envs  = (const float*)d_in[7];
  const float* w_enva  = (const float*)d_in[8];
  const float* b_enva  = (const float*)d_in[9];
  const float* w_pos   = (const float*)d_in[10];
  const float* b_pos   = (const float*)d_in[11];
  const float* w_oamp  = (const float*)d_in[12];
  const float* b_oamp  = (const float*)d_in[13];
  const float* w_def   = (const float*)d_in[14];
  const float* b_def   = (const float*)d_in[15];
  const float* items   = (const float*)d_in[16];
  const float* dho1_m  = (const float*)d_in[17];
  const float* dho1_d  = (const float*)d_in[18];
  const float* dho1_t  = (const float*)d_in[19];
  const float* dho1_x  = (const float*)d_in[20];
  const float* dho1_a  = (const float*)d_in[21];
  const float* dho2_m  = (const float*)d_in[22];
  const float* dho2_d  = (const float*)d_in[23];
  const float* dho2_t  = (const float*)d_in[24];
  const float* dho2_x  = (const float*)d_in[25];
  const float* dho2_a  = (const float*)d_in[26];
  // d_in[27..31] = dho3.* (unused by reference: it reuses dho2 for o3)
  const float* infl    = (const float*)d_in[32];
  const float* infl2   = (const float*)d_in[33];
  const float* dmix    = (const float*)d_in[34];
  const float* rooms   = (const float*)d_in[35];
  const float* w_rm    = (const float*)d_in[36];
  const float* b_rm    = (const float*)d_in[37];
  const float* w_dw    = (const float*)d_in[38];
  const float* b_dw    = (const float*)d_in[39];
  const float* w_mx    = (const float*)d_in[40];
  const float* b_mx    = (const float*)d_in[41];
  const float* w_ld    = (const float*)d_in[42];
  const float* b_ld    = (const float*)d_in[43];

  float* ws  = (float*)d_ws;
  float* out = (float*)d_out;
  const size_t SH = (size_t)FFTN * sizeof(float2);   // 256 KB LDS / block
  // SPECX/SPECI reuse the SPECR region; IMP reuses RESON; WET reuses BANK.
  float* specX = ws + WS_SPECR;
  float* specI = ws + WS_SPECR + (size_t)NEVENTS * FFTN * 2;
  float* imp   = ws + WS_RESON;
  float* wet   = ws + WS_BANK;

  // --- small linear layers ------------------------------------------------
  lin_small_kernel<<<NEVENTS * NGAUSS, 256, 0, stream>>>(events, w_envm, b_envm, ws + WS_ENVM, NGAUSS);
  lin_small_kernel<<<NEVENTS * NGAUSS, 256, 0, stream>>>(events, w_envs, b_envs, ws + WS_ENVS, NGAUSS);
  lin_small_kernel<<<NEVENTS * NGAUSS, 256, 0, stream>>>(events, w_enva, b_enva, ws + WS_ENVA, NGAUSS);
  lin_small_kernel<<<NEVENTS * 1,      256, 0, stream>>>(events, w_oamp, b_oamp, ws + WS_OA,   1);
  lin_small_kernel<<<NEVENTS * NROOMS, 256, 0, stream>>>(events, w_rm,   b_rm,   ws + WS_RM,   NROOMS);
  lin_small_kernel<<<NEVENTS * 2,      256, 0, stream>>>(events, w_dw,   b_dw,   ws + WS_DW,   2);
  lin_small_kernel<<<NEVENTS * 2,      256, 0, stream>>>(events, w_mx,   b_mx,   ws + WS_MX,   2);
  lin_small_kernel<<<NEVENTS * 1,      256, 0, stream>>>(events, w_ld,   b_ld,   ws + WS_LOUD, 1);

  // --- WMMA GEMM layers ---------------------------------------------------
  // to_resonance: (64x256)@(256x512)+b        -> 128 tiles -> 16 blocks
  gemm_wmma_kernel<1, 1, 0><<<16, 256, 0, stream>>>(events, w_tores, ws + WS_XRES, b_tores, 64, 512, 256);
  // deform: relu((64x256)@(256x128)+b)        -> 32 tiles  -> 4 blocks
  gemm_wmma_kernel<1, 1, 1><<<4, 256, 0, stream>>>(events, w_def, ws + WS_DDEF, b_def, 64, 128, 256);
  // env_pos: (64x256)@(256x192)+b             -> 48 tiles  -> 6 blocks
  gemm_wmma_kernel<1, 1, 0><<<6, 256, 0, stream>>>(events, w_pos, ws + WS_POS, b_pos, 64, 192, 256);

  // --- DHO resonance bank -------------------------------------------------
  dho_kernel<<<(NRES * NSAMP) / 256, 256, 0, stream>>>(
      dho1_m, dho1_d, dho1_t, dho1_x, dho1_a,
      dho2_m, dho2_d, dho2_t, dho2_x, dho2_a,
      infl, infl2, dmix, ws + WS_BANK);

  // --- big GEMM: resonances (256x32768) = xres(256x128) @ bank(128x32768) --
  gemm_wmma_kernel<0, 0, 0><<<4096, 256, 0, stream>>>(ws + WS_XRES, ws + WS_BANK,
                                                      ws + WS_RESON, nullptr,
                                                      256, NSAMP, NRES);

  // --- deform softmax weights / envelope+noise ----------------------------
  deform_post_kernel<<<NEVENTS, 128, 0, stream>>>(ws + WS_DDEF, items, ws + WS_DSM);
  envelope_kernel<<<NEVENTS, 256, 0, stream>>>(ws + WS_ENVM, ws + WS_ENVS,
                                               ws + WS_ENVA, ws + WS_POS,
                                               ws + WS_OA, noise, ws + WS_NTIME);

  // --- spectral convolution: resonances (*) envelope-noise ----------------
  fft_fwd_kernel<<<NEVENTS, 1024, SH, stream>>>(ws + WS_NTIME, NSAMP, NSAMP,
                                                (float2*)(ws + WS_SPECN), FFTN);
  fft_fwd_kernel<<<NEVENTS * EXPRN, 1024, SH, stream>>>(ws + WS_RESON, NSAMP, NSAMP,
                                                        (float2*)(ws + WS_SPECR), FFTN);
  fft_mulinv_kernel<<<NEVENTS * EXPRN, 1024, SH, stream>>>(
      (float2*)(ws + WS_SPECR), FFTN, (float2*)(ws + WS_SPECN), FFTN, EXPRN,
      ws + WS_RESON, NSAMP);

  // --- combine into dry signal -------------------------------------------
  combine_kernel<<<(NEVENTS * NSAMP) / 256, 256, 0, stream>>>(
      ws + WS_RESON, ws + WS_DSM, ws + WS_NTIME, ws + WS_MX, ws + WS_LOUD,
      ws + WS_XTIME);

  // --- room reverb: x (*) impulse ----------------------------------------
  impulse_kernel<<<(NEVENTS * NSAMP) / 256, 256, 0, stream>>>(ws + WS_RM, rooms, imp);
  fft_fwd_kernel<<<NEVENTS, 1024, SH, stream>>>(ws + WS_XTIME, NSAMP, NSAMP,
                                                (float2*)specX, FFTN);
  fft_fwd_kernel<<<NEVENTS, 1024, SH, stream>>>(imp, NSAMP, NSAMP,
                                                (float2*)specI, FFTN);
  fft_mulinv_kernel<<<NEVENTS, 1024, SH, stream>>>(
      (float2*)specX, FFTN, (float2*)specI, FFTN, 1, wet, NSAMP);

  // --- dry/wet blend -> output -------------------------------------------
  final_kernel<<<(NEVENTS * NSAMP) / 256, 256, 0, stream>>>(ws + WS_XTIME, wet,
                                                            ws + WS_DW, out);
}